// Projector2D_34720515620931
// MI455X (gfx1250) — compile-verified
//
#include <hip/hip_runtime.h>
#include <hip/hip_bf16.h>

// Nearest-neighbor 2x upsample, NHWC fp32.
//   in : x (8,128,128,256)  -> 33,554,432 floats (128 MiB)
//   out:   (8,256,256,256)  -> 134,217,728 floats (512 MiB)
// Pure streaming op: read each input float4 once (global_load_b128, TH=NT),
// replicate to 4 output pixels (global_store_b128, TH=NT). The 4 store
// addresses differ by +1KiB / +256KiB / +257KiB which fold into the 24-bit
// instruction offset, so one VGPR address pair serves all 4 stores.
// HBM floor: (128 + 512) MiB / 23.3 TB/s ~= 27 us.

typedef float v4f __attribute__((ext_vector_type(4)));

#define B_DIM 8
#define H_DIM 128
#define W_DIM 128
#define C_DIM 256
#define C4 (C_DIM / 4)  // 64 float4 per pixel

__global__ __launch_bounds__(256) void upsample_nn_nhwc_kernel(
    const float* __restrict__ x,
    const int* __restrict__ factor_p,
    float* __restrict__ out,
    int n4)  // number of input float4 elements
{
    int flat = blockIdx.x * 256 + threadIdx.x;  // input float4 index
    if (flat >= n4) return;

    const v4f* __restrict__ x4  = (const v4f*)x;
    v4f* __restrict__       out4 = (v4f*)out;

    // Input is contiguous in exactly (b,h,w,c) order -> one b128 NT load.
    v4f v = __builtin_nontemporal_load(&x4[flat]);

    // Decompose flat -> (b, h, w, c4). All dims are powers of two.
    int c4  = flat & (C4 - 1);
    int pix = flat >> 6;            // log2(C4) = 6
    int w   = pix & (W_DIM - 1);
    int h   = (pix >> 7) & (H_DIM - 1);
    int b   = pix >> 14;

    int f = factor_p[0];            // uniform scalar load (s_load), f==2 in harness

    if (f == 2) {
        // Ho = Wo = 256. Output float4 index of (b, 2h, 2w, c4):
        //   base4 = ((b*256 + 2h)*256 + 2w)*64 + c4
        size_t base4 = ((((size_t)b * 256 + 2 * (size_t)h) * 256) + 2 * (size_t)w) * 64
                     + (size_t)c4;
        const size_t dw4 = 64;            // +C floats  (= +1 KiB)
        const size_t dh4 = 256 * 64;      // +Wo*C floats (= +256 KiB)
        __builtin_nontemporal_store(v, &out4[base4]);
        __builtin_nontemporal_store(v, &out4[base4 + dw4]);
        __builtin_nontemporal_store(v, &out4[base4 + dh4]);
        __builtin_nontemporal_store(v, &out4[base4 + dh4 + dw4]);
    } else {
        // Generic factor fallback (correct for any f >= 1).
        size_t Wo   = (size_t)W_DIM * (size_t)f;
        size_t row4 = Wo * C4;            // float4 per output row
        size_t base4 = (((size_t)b * H_DIM * f + (size_t)h * f) * Wo
                        + (size_t)w * f) * C4 + (size_t)c4;
        for (int dh = 0; dh < f; ++dh) {
            size_t r = base4 + (size_t)dh * row4;
            for (int dw = 0; dw < f; ++dw) {
                __builtin_nontemporal_store(v, &out4[r + (size_t)dw * C4]);
            }
        }
    }
}

extern "C" void kernel_launch(void* const* d_in, const int* in_sizes, int n_in,
                              void* d_out, int out_size, void* d_ws, size_t ws_size,
                              hipStream_t stream) {
    const float* x      = (const float*)d_in[0];
    const int*   factor = (const int*)d_in[1];   // single-element device scalar
    float*       out    = (float*)d_out;

    int n4 = in_sizes[0] / 4;                    // 8,388,608 float4 work items
    int blocks = (n4 + 255) / 256;               // 32,768 blocks x 8 wave32

    upsample_nn_nhwc_kernel<<<blocks, 256, 0, stream>>>(x, factor, out, n4);
}